// DigitCaps_13142599926426
// MI455X (gfx1250) — compile-verified
//
#include <hip/hip_runtime.h>
#include <hip/hip_bf16.h>
#include <math.h>

// ---- problem constants (DigitCaps routing) ----
#define BATCH    128
#define IN_CAPS  2048
#define IN_DIM   8
#define NUM_CAPS 32
#define CAP_DIM  16
#define EPS      1e-8f

// split-K factor for the two big GEMM passes (occupancy: 256 -> 8192 waves)
#define SPLITK   32
#define KCHUNK   (IN_CAPS / SPLITK)          // 64 input capsules per chunk
#define SVEC     (BATCH * NUM_CAPS * CAP_DIM) // 65536 floats per partial slice

typedef float v2f __attribute__((ext_vector_type(2)));
typedef float v8f __attribute__((ext_vector_type(8)));

// C/D 16x16 f32 tile layout (wave32):
//   VGPR r, lanes 0-15 : M = r,     N = lane
//   VGPR r, lanes 16-31: M = r + 8, N = lane - 16
// A 16x4 f32: lane%16 = M, VGPR v holds K = v + 2*(lane/16)
// B 4x16 f32: lane%16 = N, VGPR v holds K = v + 2*(lane/16)

// -------- Pass 1 (split-K): partial s1 tiles, no scaling yet --------
// grid = (NUM_CAPS, SPLITK); 8 waves = 8 M-tiles of 16.
__global__ __launch_bounds__(256) void caps_s1_partial(const float* __restrict__ x,
                                                       const float* __restrict__ W,
                                                       float* __restrict__ part)
{
    const int j    = blockIdx.x;
    const int kc   = blockIdx.y;
    const int wav  = threadIdx.x >> 5;
    const int lane = threadIdx.x & 31;
    const int lo   = lane & 15;
    const int hi   = lane >> 4;
    const int m0   = wav * 16;
    const int i0   = kc * KCHUNK;

    const float* xA = x + (size_t)(m0 + lo) * (IN_CAPS * IN_DIM);
    const float* wB = W + (size_t)j * (CAP_DIM * IN_DIM) + lo * IN_DIM;

    v8f acc = {};
    for (int i = i0; i < i0 + KCHUNK; ++i) {
        const float* xi = xA + i * IN_DIM;
        const float* wi = wB + (size_t)i * (NUM_CAPS * CAP_DIM * IN_DIM);
#pragma unroll
        for (int s = 0; s < 2; ++s) {
            const int k = s * 4 + 2 * hi;           // e index within IN_DIM=8
            v2f a, b;
            a.x = xi[k];     a.y = xi[k + 1];
            b.x = wi[k];     b.y = wi[k + 1];
            acc = __builtin_amdgcn_wmma_f32_16x16x4_f32(
                      false, a, false, b, (short)0, acc, false, false);
        }
    }
    float* p = part + (size_t)kc * SVEC + j * CAP_DIM + lo;
#pragma unroll
    for (int r = 0; r < 8; ++r)
        p[(size_t)(m0 + r + 8 * hi) * (NUM_CAPS * CAP_DIM)] = acc[r];
}

// -------- Pass 3 (split-K): partial s2 tiles with per-row c2 scaling --------
__global__ __launch_bounds__(256) void caps_s2_partial(const float* __restrict__ x,
                                                       const float* __restrict__ W,
                                                       const float* __restrict__ c2,
                                                       float* __restrict__ part)
{
    const int j    = blockIdx.x;
    const int kc   = blockIdx.y;
    const int wav  = threadIdx.x >> 5;
    const int lane = threadIdx.x & 31;
    const int lo   = lane & 15;
    const int hi   = lane >> 4;
    const int m0   = wav * 16;
    const int i0   = kc * KCHUNK;

    const float* xA = x  + (size_t)(m0 + lo) * (IN_CAPS * IN_DIM);
    const float* cA = c2 + (size_t)(m0 + lo) * (IN_CAPS * NUM_CAPS) + j;
    const float* wB = W  + (size_t)j * (CAP_DIM * IN_DIM) + lo * IN_DIM;

    v8f acc = {};
    for (int i = i0; i < i0 + KCHUNK; ++i) {
        const float  c  = cA[(size_t)i * NUM_CAPS];
        const float* xi = xA + i * IN_DIM;
        const float* wi = wB + (size_t)i * (NUM_CAPS * CAP_DIM * IN_DIM);
#pragma unroll
        for (int s = 0; s < 2; ++s) {
            const int k = s * 4 + 2 * hi;
            v2f a, b;
            a.x = c * xi[k];   a.y = c * xi[k + 1];
            b.x = wi[k];       b.y = wi[k + 1];
            acc = __builtin_amdgcn_wmma_f32_16x16x4_f32(
                      false, a, false, b, (short)0, acc, false, false);
        }
    }
    float* p = part + (size_t)kc * SVEC + j * CAP_DIM + lo;
#pragma unroll
    for (int r = 0; r < 8; ++r)
        p[(size_t)(m0 + r + 8 * hi) * (NUM_CAPS * CAP_DIM)] = acc[r];
}

// -------- Deterministic split-K reduce + squash --------
// Thread t maps to element (b, j, d) with d = t & 15, so each 16-lane group is
// exactly one capsule vector; squash norm via xor-butterfly. Fully coalesced.
__global__ __launch_bounds__(256) void caps_reduce_squash(const float* __restrict__ part,
                                                          float* __restrict__ out,
                                                          float pre)
{
    const int t = blockIdx.x * 256 + threadIdx.x;   // 0 .. SVEC-1
    float s = 0.0f;
    for (int k = 0; k < SPLITK; ++k)                // fixed order -> deterministic
        s += part[(size_t)k * SVEC + t];
    s *= pre;
    float sq = s * s;
    sq += __shfl_xor(sq, 1, 16);
    sq += __shfl_xor(sq, 2, 16);
    sq += __shfl_xor(sq, 4, 16);
    sq += __shfl_xor(sq, 8, 16);
    const float scale = sq / ((1.0f + sq) * (sqrtf(sq) + EPS));
    out[t] = s * scale;
}

// -------- Pass 2: agreement b[b,i,j] = <u_hat[b,i,j,:], v1[b,j,:]> --------
// grid = (IN_CAPS, NUM_CAPS); 8 waves = 8 M-tiles. K=8 -> 2 WMMAs per wave.
__global__ __launch_bounds__(256) void caps_agree_kernel(const float* __restrict__ x,
                                                         const float* __restrict__ W,
                                                         const float* __restrict__ v1,
                                                         float* __restrict__ bagr)
{
    const int i    = blockIdx.x;
    const int j    = blockIdx.y;
    const int wav  = threadIdx.x >> 5;
    const int lane = threadIdx.x & 31;
    const int lo   = lane & 15;
    const int hi   = lane >> 4;
    const int m0   = wav * 16;

    const float* xi = x + (size_t)(m0 + lo) * (IN_CAPS * IN_DIM) + i * IN_DIM;
    const float* wi = W + (size_t)i * (NUM_CAPS * CAP_DIM * IN_DIM)
                        + (size_t)j * (CAP_DIM * IN_DIM) + lo * IN_DIM;

    v8f acc = {};
#pragma unroll
    for (int s = 0; s < 2; ++s) {
        const int k = s * 4 + 2 * hi;
        v2f a, b;
        a.x = xi[k];     a.y = xi[k + 1];
        b.x = wi[k];     b.y = wi[k + 1];
        acc = __builtin_amdgcn_wmma_f32_16x16x4_f32(
                  false, a, false, b, (short)0, acc, false, false);
    }
    // acc[r] = u_hat[m0 + r + 8*hi, d=lo]; dot against v1 and reduce over d.
#pragma unroll
    for (int r = 0; r < 8; ++r) {
        const int m = m0 + r + 8 * hi;
        float t = acc[r] * v1[(size_t)m * (NUM_CAPS * CAP_DIM) + j * CAP_DIM + lo];
        t += __shfl_xor(t, 1, 16);
        t += __shfl_xor(t, 2, 16);
        t += __shfl_xor(t, 4, 16);
        t += __shfl_xor(t, 8, 16);
        if (lo == r)
            bagr[(size_t)m * (IN_CAPS * NUM_CAPS) + (size_t)i * NUM_CAPS + j] = t;
    }
}

// -------- Softmax over j (32 values) in place: b_agr -> c2 --------
__global__ __launch_bounds__(256) void caps_softmax_kernel(float* __restrict__ bagr)
{
    const size_t row = (size_t)blockIdx.x * blockDim.x + threadIdx.x; // (b, i)
    float* p = bagr + row * NUM_CAPS;
    float vals[NUM_CAPS];
    float mx = -INFINITY;
#pragma unroll
    for (int j = 0; j < NUM_CAPS; ++j) { vals[j] = p[j]; mx = fmaxf(mx, vals[j]); }
    float sum = 0.0f;
#pragma unroll
    for (int j = 0; j < NUM_CAPS; ++j) { vals[j] = __expf(vals[j] - mx); sum += vals[j]; }
    const float inv = 1.0f / sum;
#pragma unroll
    for (int j = 0; j < NUM_CAPS; ++j) p[j] = vals[j] * inv;
}

extern "C" void kernel_launch(void* const* d_in, const int* in_sizes, int n_in,
                              void* d_out, int out_size, void* d_ws, size_t ws_size,
                              hipStream_t stream)
{
    const float* x = (const float*)d_in[0];   // [128, 2048, 8]
    const float* W = (const float*)d_in[1];   // [1, 2048, 32, 16, 8]
    float* vout = (float*)d_out;              // [128, 32, 16]

    float* v1   = (float*)d_ws;                               // 65,536 floats
    float* bagr = v1 + (size_t)SVEC;                          // 8,388,608 floats
    float* part = bagr + (size_t)BATCH * IN_CAPS * NUM_CAPS;  // SPLITK * 65,536 floats

    // iter 1: s1 partials -> reduce (x 1/32 uniform coupling) + squash -> v1
    caps_s1_partial   <<<dim3(NUM_CAPS, SPLITK), 256, 0, stream>>>(x, W, part);
    caps_reduce_squash<<<dim3(SVEC / 256),       256, 0, stream>>>(part, v1,
                                                                  1.0f / (float)NUM_CAPS);
    // agreement + softmax -> c2 (in place in bagr)
    caps_agree_kernel  <<<dim3(IN_CAPS, NUM_CAPS),       256, 0, stream>>>(x, W, v1, bagr);
    caps_softmax_kernel<<<dim3((BATCH * IN_CAPS) / 256), 256, 0, stream>>>(bagr);
    // iter 2: s2 partials (c2-weighted) -> reduce + squash -> v (output)
    caps_s2_partial   <<<dim3(NUM_CAPS, SPLITK), 256, 0, stream>>>(x, W, bagr, part);
    caps_reduce_squash<<<dim3(SVEC / 256),       256, 0, stream>>>(part, vout, 1.0f);
}